// PretrainModel_68410239091019
// MI455X (gfx1250) — compile-verified
//
#include <hip/hip_runtime.h>
#include <math.h>

#define N_BATCH 1024
#define N_NEG   64
#define N_ROWS  2112      // 1024 inf + 1024 lf + 64 nf
#define K_RPR   128
#define FEAT    128
#define NVEC    200
#define M_TILES 132       // 2112 / 16
#define N_TILES 13        // ceil(200 / 16)
#define N_TILES_TOTAL (M_TILES * N_TILES)   // 1716
#define WAVES_PER_BLOCK 8

typedef __attribute__((ext_vector_type(2))) float v2f;
typedef __attribute__((ext_vector_type(8))) float v8f;

__device__ __forceinline__ float log_sig(float x) {
  // log(sigmoid(x) + 0.001)
  return logf(1.0f / (1.0f + expf(-x)) + 0.001f);
}

// ---------------------------------------------------------------------------
// Kernel 1: weighted neighbor aggregation.
// One block per output row (2112), 128 threads = one feature column each.
// agg[b][f] = sum_k rpr_matrix[node][k] * features[rpr_arg[node][k]][f]
// features (51 MB) is L2-resident (192 MB L2), so gathers hit L2.
// ---------------------------------------------------------------------------
__global__ __launch_bounds__(128) void aggregate_kernel(
    const int* __restrict__ ti, const int* __restrict__ tl,
    const int* __restrict__ ns,
    const int* __restrict__ rpr_arg, const float* __restrict__ rpr_mat,
    const float* __restrict__ features, float* __restrict__ agg) {
  int b = blockIdx.x;
  int f = threadIdx.x;
  int node;
  if (b < N_BATCH)          node = ti[b];
  else if (b < 2 * N_BATCH) node = tl[b - N_BATCH];
  else                      node = ns[b - 2 * N_BATCH];

  __shared__ int   s_nbr[K_RPR];
  __shared__ float s_w[K_RPR];
  s_nbr[f] = rpr_arg[(size_t)node * K_RPR + f];
  s_w[f]   = rpr_mat[(size_t)node * K_RPR + f];
  __syncthreads();

  float acc = 0.0f;
#pragma unroll 8
  for (int k = 0; k < K_RPR; ++k) {
    // prefetch the feature row 8 neighbors ahead (global_prefetch_b8)
    int pk = (k + 8) & (K_RPR - 1);
    __builtin_prefetch(features + (size_t)s_nbr[pk] * FEAT + f, 0, 1);
    acc = fmaf(s_w[k], features[(size_t)s_nbr[k] * FEAT + f], acc);
  }
  agg[(size_t)b * FEAT + f] = acc;
}

// ---------------------------------------------------------------------------
// Kernel 2: out = relu(agg[2112x128] @ agg_W[128x200]) via fp32 WMMA 16x16x4.
// 8 waves per block, one 16x16 tile per wave; 32 K-steps of 4.
// A layout (16x4 f32): lane L<16 holds A[M=L][K=k0],A[L][k0+1]; lanes 16-31
// hold K=k0+2,k0+3. B (4x16) mirrors with N = lane%16. C/D: VGPR i = row
// m0+i (lanes 0-15) / m0+i+8 (lanes 16-31), col = n0 + lane%16.
//
// Columns >= NVEC: WMMA output column n depends only on B column n, so
// out-of-range lanes just CLAMP their load address (branch-free) — their
// garbage products land in output columns that are never stored.
// ---------------------------------------------------------------------------
__global__ __launch_bounds__(WAVES_PER_BLOCK * 32) void gemm_relu_wmma(
    const float* __restrict__ agg, const float* __restrict__ W,
    float* __restrict__ out) {
  int lane = threadIdx.x & 31;
  int wave = threadIdx.x >> 5;
  int tile = blockIdx.x * WAVES_PER_BLOCK + wave;   // wave-uniform
  if (tile >= N_TILES_TOTAL) return;                // whole-wave exit: WMMA
                                                    // never runs w/ EXEC!=~0
  int tm = tile % M_TILES;
  int tn = tile / M_TILES;
  int m0 = tm * 16;
  int n0 = tn * 16;

  int r  = lane & 15;
  int kb = (lane >> 4) * 2;

  const float* arow = agg + (size_t)(m0 + r) * FEAT;   // m0+r < 2112 always
  int  col  = n0 + r;
  int  colc = col < NVEC ? col : (NVEC - 1);           // clamped, always legal
  const float* bcol = W + colc;

  v8f c = {};
#pragma unroll
  for (int k0 = 0; k0 < FEAT; k0 += 4) {
    v2f a, bfr;
    a.x   = arow[k0 + kb];
    a.y   = arow[k0 + kb + 1];
    bfr.x = bcol[(size_t)(k0 + kb) * NVEC];
    bfr.y = bcol[(size_t)(k0 + kb + 1) * NVEC];
    c = __builtin_amdgcn_wmma_f32_16x16x4_f32(
        /*neg_a=*/false, a, /*neg_b=*/false, bfr,
        /*c_mod=*/(short)0, c, /*reuse_a=*/false, /*reuse_b=*/false);
  }

  if (col < NVEC) {
    int row_half = (lane >> 4) * 8;
#pragma unroll
    for (int i = 0; i < 8; ++i) {
      int row = m0 + row_half + i;
      float v = c[i];
      out[(size_t)row * NVEC + col] = v > 0.0f ? v : 0.0f;
    }
  }
}

// ---------------------------------------------------------------------------
// Kernel 3: per-row terms p1,p3,p5,p7 (200-dim dots), one wave32 per row.
// row_terms[b] = rho1*ls(inf.lf) + rho2*ls(em.tw) + rho3*(ls(em.lf)+ls(tw.inf))
// ---------------------------------------------------------------------------
__global__ __launch_bounds__(256) void row_terms_kernel(
    const int* __restrict__ ti, const int* __restrict__ tl,
    const float* __restrict__ embeddings, const float* __restrict__ nce_w,
    const float* __restrict__ outvec, float* __restrict__ row_terms) {
  int lane = threadIdx.x & 31;
  int b = blockIdx.x * 8 + (threadIdx.x >> 5);
  if (b >= N_BATCH) return;

  const float* inf = outvec + (size_t)b * NVEC;
  const float* lf  = outvec + (size_t)(N_BATCH + b) * NVEC;
  const float* em  = embeddings + (size_t)ti[b] * NVEC;
  const float* tw  = nce_w + (size_t)tl[b] * NVEC;

  float d1 = 0, d3 = 0, d5 = 0, d7 = 0;
  for (int j = lane; j < NVEC; j += 32) {
    float vi = inf[j], vl = lf[j], ve = em[j], vt = tw[j];
    d1 = fmaf(vi, vl, d1);
    d3 = fmaf(ve, vt, d3);
    d5 = fmaf(ve, vl, d5);
    d7 = fmaf(vt, vi, d7);
  }
#pragma unroll
  for (int m = 16; m; m >>= 1) {
    d1 += __shfl_xor(d1, m, 32);
    d3 += __shfl_xor(d3, m, 32);
    d5 += __shfl_xor(d5, m, 32);
    d7 += __shfl_xor(d7, m, 32);
  }
  if (lane == 0)
    row_terms[b] = 1.5f * log_sig(d1) + 0.75f * log_sig(d3) +
                   1.5f * log_sig(d5) + 1.5f * log_sig(d7);
}

// ---------------------------------------------------------------------------
// Kernel 4: column sums for the factorized full-sum terms.
// sum(A @ B.T) == (colsum-over-rows A) . (colsum-over-rows B)  (exact)
// colsums layout: [0]=S_inf [1]=S_nf [2]=S_embed [3]=S_truew [4]=S_falsew
// ---------------------------------------------------------------------------
__global__ __launch_bounds__(256) void col_sums_kernel(
    const int* __restrict__ ti, const int* __restrict__ tl,
    const int* __restrict__ ns,
    const float* __restrict__ embeddings, const float* __restrict__ nce_w,
    const float* __restrict__ outvec, float* __restrict__ colsums) {
  int j = threadIdx.x;
  if (j >= NVEC) return;
  float s_inf = 0, s_nf = 0, s_em = 0, s_tw = 0, s_fw = 0;
  for (int b = 0; b < N_BATCH; ++b) {
    s_inf += outvec[(size_t)b * NVEC + j];
    s_em  += embeddings[(size_t)ti[b] * NVEC + j];
    s_tw  += nce_w[(size_t)tl[b] * NVEC + j];
  }
  for (int n = 0; n < N_NEG; ++n) {
    s_nf += outvec[(size_t)(2 * N_BATCH + n) * NVEC + j];
    s_fw += nce_w[(size_t)ns[n] * NVEC + j];
  }
  colsums[0 * NVEC + j] = s_inf;
  colsums[1 * NVEC + j] = s_nf;
  colsums[2 * NVEC + j] = s_em;
  colsums[3 * NVEC + j] = s_tw;
  colsums[4 * NVEC + j] = s_fw;
}

// ---------------------------------------------------------------------------
// Kernel 5: final scalar — p2/p4/p6/p8 dots + row-term sum -> loss.
// ---------------------------------------------------------------------------
__global__ __launch_bounds__(32) void final_kernel(
    const float* __restrict__ colsums, const float* __restrict__ row_terms,
    float* __restrict__ out) {
  int lane = threadIdx.x;
  float d2 = 0, d4 = 0, d6 = 0, d8 = 0, rsum = 0;
  for (int j = lane; j < NVEC; j += 32) {
    float si = colsums[j];
    float sn = colsums[NVEC + j];
    float se = colsums[2 * NVEC + j];
    float st = colsums[3 * NVEC + j];
    float sf = colsums[4 * NVEC + j];
    d2 = fmaf(si, sn, d2);   // sum(inf @ nf.T)
    d4 = fmaf(se, sf, d4);   // sum(embed @ false_w.T)
    d6 = fmaf(se, sn, d6);   // sum(embed @ nf.T)
    d8 = fmaf(st, sn, d8);   // sum(true_w @ nf.T)
  }
  for (int b = lane; b < N_BATCH; b += 32) rsum += row_terms[b];
#pragma unroll
  for (int m = 16; m; m >>= 1) {
    d2 += __shfl_xor(d2, m, 32);
    d4 += __shfl_xor(d4, m, 32);
    d6 += __shfl_xor(d6, m, 32);
    d8 += __shfl_xor(d8, m, 32);
    rsum += __shfl_xor(rsum, m, 32);
  }
  if (lane == 0) {
    float scal = 1.5f * log_sig(-d2) + 0.75f * log_sig(-d4) +
                 1.5f * log_sig(-d6) + 1.5f * log_sig(-d8);
    float total = rsum + (float)N_BATCH * scal;
    out[0] = -total / (float)N_BATCH;
  }
}

// ---------------------------------------------------------------------------
extern "C" void kernel_launch(void* const* d_in, const int* in_sizes, int n_in,
                              void* d_out, int out_size, void* d_ws,
                              size_t ws_size, hipStream_t stream) {
  (void)in_sizes; (void)n_in; (void)out_size; (void)ws_size;
  const int*   ti         = (const int*)d_in[0];
  const int*   tl         = (const int*)d_in[1];
  const int*   ns         = (const int*)d_in[2];
  const int*   rpr_arg    = (const int*)d_in[3];
  const float* rpr_mat    = (const float*)d_in[4];
  const float* features   = (const float*)d_in[5];
  const float* embeddings = (const float*)d_in[6];
  const float* nce_w      = (const float*)d_in[7];
  const float* agg_W      = (const float*)d_in[8];
  float*       out        = (float*)d_out;

  // workspace layout (floats): agg | outvec | colsums | row_terms  (~2.8 MB)
  float* ws        = (float*)d_ws;
  float* agg       = ws;                                 // 2112*128
  float* outvec    = agg + (size_t)N_ROWS * FEAT;        // 2112*200
  float* colsums   = outvec + (size_t)N_ROWS * NVEC;     // 5*200
  float* row_terms = colsums + 5 * NVEC;                 // 1024

  aggregate_kernel<<<N_ROWS, 128, 0, stream>>>(ti, tl, ns, rpr_arg, rpr_mat,
                                               features, agg);
  int gemm_blocks = (N_TILES_TOTAL + WAVES_PER_BLOCK - 1) / WAVES_PER_BLOCK;
  gemm_relu_wmma<<<gemm_blocks, WAVES_PER_BLOCK * 32, 0, stream>>>(agg, agg_W,
                                                                   outvec);
  row_terms_kernel<<<N_BATCH / 8, 256, 0, stream>>>(ti, tl, embeddings, nce_w,
                                                    outvec, row_terms);
  col_sums_kernel<<<1, 256, 0, stream>>>(ti, tl, ns, embeddings, nce_w, outvec,
                                         colsums);
  final_kernel<<<1, 32, 0, stream>>>(colsums, row_terms, out);
}